// Ngram_24618752540965
// MI455X (gfx1250) — compile-verified
//
#include <hip/hip_runtime.h>
#include <stdint.h>

// N-gram logits: out[p,v] = bias[v] + sum_k W[v][k*(V+1) + ctx_k(p)]
// Memory-bound gather problem (0.27 GFLOP vs ~1 GB traffic @ 23.3 TB/s ~ 40us).
// Strategy:
//   1) TDM-tiled transpose of W (V x FAN) -> WT (FAN x V) in workspace:
//      one tensor_load_to_lds per 64x64 tile (strided 2-D tile, 256B rows,
//      64KB pitch), TDM pad_interval=64/pad_amount=1 -> 65-dword LDS pitch,
//      bank-conflict-free transposed reads, coalesced 256B stores.
//   2) TDM gather-mode load of the 4 needed WT rows (64KB) into LDS per
//      position (row indices k*4097+ctx_k are strictly increasing, as the
//      ISA requires for correct OOB), then 8 waves sum rows + bias with
//      float4 LDS reads and coalesced float4 stores.

#define VOCAB 4096
#define MORD  4
#define BB    8
#define LL    2048
#define FAN   (MORD * (VOCAB + 1))   // 16388
#define NPOS  (BB * LL)              // 16384

typedef __attribute__((ext_vector_type(4))) unsigned int v4u;
typedef __attribute__((ext_vector_type(8))) int          v8i;
typedef __attribute__((ext_vector_type(4))) int          v4i;

// ---- TDM descriptor builders (CDNA5 ISA, async/tensor ch. 8) ----
__device__ __forceinline__ v4u tdm_group0(unsigned lds_addr,
                                          unsigned long long gaddr,
                                          bool gather_mode) {
  v4u g;
  // [1:0] count=1 (valid), [30] gather_index_size=0 (16-bit), [31] gather_mode
  g.x = 1u | (gather_mode ? (1u << 31) : 0u);
  g.y = lds_addr;                                               // [63:32] lds_addr
  g.z = (unsigned)(gaddr & 0xFFFFFFFFull);                      // [95:64] global_addr lo
  g.w = (unsigned)((gaddr >> 32) & 0x1FFFFFFull) | (2u << 30);  // addr[56:32] | type=2
  return g;
}

__device__ __forceinline__ v8i tdm_group1(unsigned flags_w0, unsigned dim0,
                                          unsigned dim1, unsigned tile0,
                                          unsigned tile1_tile2,
                                          unsigned long long stride0) {
  v8i g;
  g[0] = (int)flags_w0;                                       // wg_mask=0|data_size|pad cfg
  g[1] = (int)((dim0 & 0xFFFFu) << 16);                       // abar_addr=0 | tensor_dim0[15:0]
  g[2] = (int)((dim0 >> 16) | ((dim1 & 0xFFFFu) << 16));      // dim0[31:16] | dim1[15:0]
  g[3] = (int)((dim1 >> 16) | (tile0 << 16));                 // dim1[31:16] | tile_dim0
  g[4] = (int)tile1_tile2;                                    // tile_dim1 | tile_dim2<<16
  g[5] = (int)(unsigned)(stride0 & 0xFFFFFFFFull);            // tensor_dim0_stride[31:0]
  g[6] = (int)(unsigned)((stride0 >> 32) & 0xFFFFull);        // stride0[47:32] | stride1 lo=0
  g[7] = 0;
  return g;
}

// ---- Phase 1: W (V x FAN, row major) -> WT (FAN x V, row major) ----
// One TDM strided-tile load (64 rows x 256B, row pitch FAN*4) per 64x64 tile,
// with pad_interval=64 dwords (code 5) / pad_amount=1 dword -> LDS pitch 65.
#define TT 64
__global__ __launch_bounds__(256) void Ngram_transpose(const float* __restrict__ W,
                                                       float* __restrict__ WT) {
  __shared__ float tile[TT * (TT + 1)];    // 65-dword pitch; padding done by TDM
  const int v0 = blockIdx.x * TT;          // W row / WT column tile origin
  const int r0 = blockIdx.y * TT;          // W col / WT row tile origin

  if (threadIdx.x < 32) {                  // wave 0 issues the DMA
    unsigned long long gaddr = (unsigned long long)(uintptr_t)W +
        ((unsigned long long)v0 * FAN + (unsigned long long)r0) * 4ull;
    v4u g0 = tdm_group0((unsigned)(uintptr_t)tile, gaddr, /*gather=*/false);
    // data_size=4B (2<<16) | pad_enable (1<<20) | pad_interval=64 dwords (5<<22)
    // | pad_amount=1 dword (0<<25)
    const unsigned flags = (2u << 16) | (1u << 20) | (5u << 22);
    v8i g1 = tdm_group1(flags,
                        (unsigned)(FAN - r0),        // remaining extent dim0 (OOB->0)
                        (unsigned)(VOCAB - v0),      // remaining extent dim1
                        (unsigned)TT,                // tile_dim0 (contiguous, r)
                        (unsigned)TT,                // tile_dim1 (rows, v); tile_dim2=0
                        (unsigned long long)FAN);    // row pitch in elements
    v4i z4 = {0, 0, 0, 0};
    v8i z8 = {0, 0, 0, 0, 0, 0, 0, 0};
    __builtin_amdgcn_tensor_load_to_lds(g0, g1, z4, z4, z8, 0);
    __builtin_amdgcn_s_wait_tensorcnt(0);
  }
  __syncthreads();

  const int tx = threadIdx.x & (TT - 1);   // v offset within tile (store-contiguous)
  const int ty = threadIdx.x >> 6;         // 0..3
#pragma unroll
  for (int i = 0; i < 16; ++i) {
    const int x = ty + i * 4;              // r offset within tile
    const int r = r0 + x;
    if (r < FAN) {                         // FAN=16388 isn't a multiple of 64
      // LDS pitch 65: lanes tx consecutive -> addr tx*65+x, 65 % 64 == 1 ->
      // all 64 banks distinct, conflict-free. Stores: 256B contiguous per row.
      WT[(unsigned long long)r * VOCAB + (unsigned)(v0 + tx)] = tile[tx * (TT + 1) + x];
    }
  }
}

// ---- Phase 2: per position, TDM gather of 4 WT rows -> LDS, then sum ----
__global__ __launch_bounds__(256) void Ngram_gather(const int* __restrict__ idx,
                                                    const float* __restrict__ WT,
                                                    const float* __restrict__ bias,
                                                    float* __restrict__ out) {
  __shared__ float rows[MORD * VOCAB];     // 4 x 16KB = 64KB -> ~5 blocks/WGP
  const int p  = blockIdx.x;
  const int bi = p >> 11;                  // / LL
  const int li = p & (LL - 1);

  if (threadIdx.x < 32) {                  // wave 0 issues the DMA
    unsigned r[MORD];
#pragma unroll
    for (int k = 0; k < MORD; ++k) {
      const int t = li - (MORD - 1 - k);
      int j = (t >= 0) ? idx[bi * LL + t] : VOCAB;   // pad token = VOCAB
      j = __builtin_amdgcn_readfirstlane(j);         // force SGPR for descriptor
      r[k] = (unsigned)(k * (VOCAB + 1) + j);        // strictly increasing across k
    }
    v4u g0 = tdm_group0((unsigned)(uintptr_t)rows,
                        (unsigned long long)(uintptr_t)WT, /*gather=*/true);
    v8i g1 = tdm_group1((2u << 16),                  // data_size=4B, no padding
                        (unsigned)VOCAB,             // tensor_dim0 = row length
                        (unsigned)FAN,               // tensor_dim1 = #rows
                        (unsigned)VOCAB,             // tile_dim0 = full row
                        (unsigned)MORD,              // tile_dim1 = #valid indices
                        (unsigned long long)VOCAB);  // row pitch
    v4i g2 = { (int)(r[0] | (r[1] << 16)), (int)(r[2] | (r[3] << 16)), 0, 0 };
    v4i g3 = {0, 0, 0, 0};
    v8i z8 = {0, 0, 0, 0, 0, 0, 0, 0};
    __builtin_amdgcn_tensor_load_to_lds(g0, g1, g2, g3, z8, 0);
    __builtin_amdgcn_s_wait_tensorcnt(0);
  }
  __syncthreads();

  const float4* __restrict__ b4 = (const float4*)bias;
  const float4* __restrict__ r4 = (const float4*)rows;
  float4* __restrict__ o4 = (float4*)(out + (unsigned long long)p * VOCAB);
#pragma unroll
  for (int i = 0; i < 4; ++i) {
    const int c = (int)threadIdx.x + i * 256;        // 1024 float4 chunks
    float4 a0 = r4[c];
    float4 a1 = r4[c + 1024];
    float4 a2 = r4[c + 2048];
    float4 a3 = r4[c + 3072];
    float4 s  = b4[c];
    // same accumulation order as reference: b, then tables k=0..3
    s.x += a0.x; s.y += a0.y; s.z += a0.z; s.w += a0.w;
    s.x += a1.x; s.y += a1.y; s.z += a1.z; s.w += a1.w;
    s.x += a2.x; s.y += a2.y; s.z += a2.z; s.w += a2.w;
    s.x += a3.x; s.y += a3.y; s.z += a3.z; s.w += a3.w;
    o4[c] = s;
  }
}

// ---- Fallback if workspace can't hold WT (reads W columns directly) ----
__global__ __launch_bounds__(256) void Ngram_direct(const int* __restrict__ idx,
                                                    const float* __restrict__ W,
                                                    const float* __restrict__ bias,
                                                    float* __restrict__ out) {
  const int p  = blockIdx.x;
  const int bi = p >> 11;
  const int li = p & (LL - 1);
  unsigned r[MORD];
#pragma unroll
  for (int k = 0; k < MORD; ++k) {
    const int t = li - (MORD - 1 - k);
    const int j = (t >= 0) ? idx[bi * LL + t] : VOCAB;
    r[k] = (unsigned)(k * (VOCAB + 1) + j);
  }
  for (int v = (int)threadIdx.x; v < VOCAB; v += 256) {
    const float* wr = W + (unsigned long long)v * FAN;
    float s = bias[v];
    s += wr[r[0]]; s += wr[r[1]]; s += wr[r[2]]; s += wr[r[3]];
    out[(unsigned long long)p * VOCAB + v] = s;
  }
}

extern "C" void kernel_launch(void* const* d_in, const int* in_sizes, int n_in,
                              void* d_out, int out_size, void* d_ws, size_t ws_size,
                              hipStream_t stream) {
  const int*   idx  = (const int*)d_in[0];    // [B, L] int32
  const float* W    = (const float*)d_in[1];  // [V, FAN] fp32
  const float* bias = (const float*)d_in[2];  // [V] fp32
  float*       out  = (float*)d_out;          // [B, L, V] fp32

  const size_t need = (size_t)FAN * (size_t)VOCAB * sizeof(float);  // 268 MB
  if (ws_size >= need) {
    float* WT = (float*)d_ws;
    dim3 tgrid(VOCAB / TT, (FAN + TT - 1) / TT);  // 64 x 257 tiles
    Ngram_transpose<<<tgrid, 256, 0, stream>>>(W, WT);
    Ngram_gather<<<NPOS, 256, 0, stream>>>(idx, WT, bias, out);
  } else {
    Ngram_direct<<<NPOS, 256, 0, stream>>>(idx, W, bias, out);
  }
}